// GCN_72962904424636
// MI455X (gfx1250) — compile-verified
//
#include <hip/hip_runtime.h>

typedef float v2f __attribute__((ext_vector_type(2)));
typedef float v8f __attribute__((ext_vector_type(8)));

#define N_NODES 100000
#define N_EDGES 1600000
#define N_FEAT  256
#define N_HID   64
#define N_CLS   16
#define N_TILES (N_NODES / 16)   // 6250, exact

// ---------------------------------------------------------------------------
// zero-init scratch (scatter-add targets) — grid-stride float4
// ---------------------------------------------------------------------------
__global__ void gcn_zero_kernel(float4* __restrict__ p, size_t n4) {
    size_t i = (size_t)blockIdx.x * blockDim.x + threadIdx.x;
    size_t stride = (size_t)gridDim.x * blockDim.x;
    float4 z = {0.f, 0.f, 0.f, 0.f};
    for (; i < n4; i += stride) p[i] = z;
}

// ---------------------------------------------------------------------------
// GEMM1: h0[N,64] = x[N,256] @ W1[256,64] + b1   (fp32 WMMA 16x16x4)
// one wave -> one 16-row tile, full 64-wide output (4 accumulators)
// ---------------------------------------------------------------------------
__global__ void __launch_bounds__(128) gcn_gemm1_kernel(
        const float* __restrict__ x, const float* __restrict__ W1,
        const float* __restrict__ b1, float* __restrict__ h0) {
    __shared__ float sW[N_FEAT * N_HID];   // 64 KB (WGP LDS is 320 KB)
    {
        const float4* W4 = (const float4*)W1;
        float4* s4 = (float4*)sW;
        for (int i = threadIdx.x; i < (N_FEAT * N_HID) / 4; i += 128) s4[i] = W4[i];
    }
    __syncthreads();

    const int wave = threadIdx.x >> 5;
    const int lane = threadIdx.x & 31;
    const int half = lane >> 4;        // 0: lanes 0-15, 1: lanes 16-31
    const int m    = lane & 15;

    const int tile = blockIdx.x * 4 + wave;   // wave-uniform -> branch keeps EXEC all-1s
    if (tile >= N_TILES) return;

    const int row = tile * 16 + m;
    const float* __restrict__ xrow = x + (size_t)row * N_FEAT;

    v8f acc0 = {}, acc1 = {}, acc2 = {}, acc3 = {};

    for (int k0 = 0; k0 < N_FEAT; k0 += 4) {
        const int k = k0 + 2 * half;   // lanes 0-15 hold K=k0,k0+1; lanes 16-31 K=k0+2,k0+3
        v2f a;
        a.x = xrow[k];
        a.y = xrow[k + 1];
        const float* wk0 = sW + (size_t)k * N_HID;
        const float* wk1 = wk0 + N_HID;
        v2f b0, b1v, b2v, b3v;
        b0.x  = wk0[m];      b0.y  = wk1[m];
        b1v.x = wk0[16 + m]; b1v.y = wk1[16 + m];
        b2v.x = wk0[32 + m]; b2v.y = wk1[32 + m];
        b3v.x = wk0[48 + m]; b3v.y = wk1[48 + m];
        acc0 = __builtin_amdgcn_wmma_f32_16x16x4_f32(false, a, false, b0,  (short)0, acc0, false, false);
        acc1 = __builtin_amdgcn_wmma_f32_16x16x4_f32(false, a, false, b1v, (short)0, acc1, false, false);
        acc2 = __builtin_amdgcn_wmma_f32_16x16x4_f32(false, a, false, b2v, (short)0, acc2, false, false);
        acc3 = __builtin_amdgcn_wmma_f32_16x16x4_f32(false, a, false, b3v, (short)0, acc3, false, false);
    }

    // C/D layout: VGPR r -> M = half*8 + r, N = m (per lane half)
    float* __restrict__ out = h0 + (size_t)tile * 16 * N_HID;
    const float bb0 = b1[m], bb1 = b1[16 + m], bb2 = b1[32 + m], bb3 = b1[48 + m];
    #pragma unroll
    for (int r = 0; r < 8; ++r) {
        const size_t orow = (size_t)(half * 8 + r) * N_HID;
        out[orow + m]      = acc0[r] + bb0;
        out[orow + 16 + m] = acc1[r] + bb1;
        out[orow + 32 + m] = acc2[r] + bb2;
        out[orow + 48 + m] = acc3[r] + bb3;
    }
}

// ---------------------------------------------------------------------------
// SpMM, D=64: out[row] += vals[e] * h[col]   (edge x feat parallel, atomics)
// ---------------------------------------------------------------------------
__global__ void __launch_bounds__(256) gcn_spmm64_kernel(
        const int* __restrict__ arow, const int* __restrict__ acol,
        const float* __restrict__ avals, const float* __restrict__ h,
        float* __restrict__ out) {
    size_t t = (size_t)blockIdx.x * 256 + threadIdx.x;
    if (t >= (size_t)N_EDGES * N_HID) return;
    const int e = (int)(t >> 6);
    const int f = (int)(t & 63);
    const int c = acol[e];
    const int r = arow[e];
    const float v = avals[e];
    atomicAdd(&out[(size_t)r * N_HID + f], v * h[(size_t)c * N_HID + f]);
}

// ---------------------------------------------------------------------------
// GEMM2: h2[N,16] = relu(h1)[N,64] @ W2[64,16] + b2   (fp32 WMMA, relu fused)
// ---------------------------------------------------------------------------
__global__ void __launch_bounds__(128) gcn_gemm2_kernel(
        const float* __restrict__ h1, const float* __restrict__ W2,
        const float* __restrict__ b2, float* __restrict__ h2) {
    __shared__ float sW[N_HID * N_CLS];   // 4 KB
    for (int i = threadIdx.x; i < N_HID * N_CLS; i += 128) sW[i] = W2[i];
    __syncthreads();

    const int wave = threadIdx.x >> 5;
    const int lane = threadIdx.x & 31;
    const int half = lane >> 4;
    const int m    = lane & 15;

    const int tile = blockIdx.x * 4 + wave;
    if (tile >= N_TILES) return;

    const int row = tile * 16 + m;
    const float* __restrict__ hrow = h1 + (size_t)row * N_HID;

    v8f acc = {};
    for (int k0 = 0; k0 < N_HID; k0 += 4) {
        const int k = k0 + 2 * half;
        v2f a;
        a.x = fmaxf(hrow[k], 0.f);        // fused ReLU
        a.y = fmaxf(hrow[k + 1], 0.f);
        v2f b;
        b.x = sW[(size_t)k * N_CLS + m];
        b.y = sW[(size_t)(k + 1) * N_CLS + m];
        acc = __builtin_amdgcn_wmma_f32_16x16x4_f32(false, a, false, b, (short)0, acc, false, false);
    }

    float* __restrict__ out = h2 + (size_t)tile * 16 * N_CLS;
    const float bb = b2[m];
    #pragma unroll
    for (int r = 0; r < 8; ++r)
        out[(size_t)(half * 8 + r) * N_CLS + m] = acc[r] + bb;
}

// ---------------------------------------------------------------------------
// SpMM, D=16
// ---------------------------------------------------------------------------
__global__ void __launch_bounds__(256) gcn_spmm16_kernel(
        const int* __restrict__ arow, const int* __restrict__ acol,
        const float* __restrict__ avals, const float* __restrict__ h,
        float* __restrict__ out) {
    size_t t = (size_t)blockIdx.x * 256 + threadIdx.x;
    if (t >= (size_t)N_EDGES * N_CLS) return;
    const int e = (int)(t >> 4);
    const int f = (int)(t & 15);
    const int c = acol[e];
    const int r = arow[e];
    const float v = avals[e];
    atomicAdd(&out[(size_t)r * N_CLS + f], v * h[(size_t)c * N_CLS + f]);
}

// ---------------------------------------------------------------------------
// log_softmax over 16 classes, one thread per node
// ---------------------------------------------------------------------------
__global__ void __launch_bounds__(256) gcn_lsm_kernel(
        const float* __restrict__ s, float* __restrict__ out) {
    const int r = blockIdx.x * 256 + threadIdx.x;
    if (r >= N_NODES) return;
    const float* __restrict__ p = s + (size_t)r * N_CLS;
    float v[N_CLS];
    float mx = -3.402823466e+38f;
    #pragma unroll
    for (int i = 0; i < N_CLS; ++i) { v[i] = p[i]; mx = fmaxf(mx, v[i]); }
    float sum = 0.f;
    #pragma unroll
    for (int i = 0; i < N_CLS; ++i) sum += __expf(v[i] - mx);
    const float lse = mx + __logf(sum);
    float* __restrict__ o = out + (size_t)r * N_CLS;
    #pragma unroll
    for (int i = 0; i < N_CLS; ++i) o[i] = v[i] - lse;
}

// ---------------------------------------------------------------------------
extern "C" void kernel_launch(void* const* d_in, const int* in_sizes, int n_in,
                              void* d_out, int out_size, void* d_ws, size_t ws_size,
                              hipStream_t stream) {
    const float* x        = (const float*)d_in[0];
    const int*   adj_row  = (const int*)  d_in[1];
    const int*   adj_col  = (const int*)  d_in[2];
    const float* adj_vals = (const float*)d_in[3];
    const float* W1       = (const float*)d_in[4];
    const float* b1       = (const float*)d_in[5];
    const float* W2       = (const float*)d_in[6];
    const float* b2       = (const float*)d_in[7];
    float* out = (float*)d_out;

    // workspace layout (fp32): h0[N,64] | h1[N,64] | h2[N,16] | s2[N,16]  ~64 MB
    float* ws = (float*)d_ws;
    float* h0 = ws;
    float* h1 = h0 + (size_t)N_NODES * N_HID;
    float* h2 = h1 + (size_t)N_NODES * N_HID;
    float* s2 = h2 + (size_t)N_NODES * N_CLS;

    // zero h1..s2 (h2 included; it's overwritten anyway) — deterministic per call
    {
        size_t nz = (size_t)N_NODES * (N_HID + N_CLS + N_CLS);  // floats
        size_t n4 = nz / 4;
        int blocks = 2048;
        gcn_zero_kernel<<<blocks, 256, 0, stream>>>((float4*)h1, n4);
    }

    // GEMM1 (WMMA f32)
    {
        int blocks = (N_TILES + 3) / 4;   // 4 waves/block, 1 tile/wave
        gcn_gemm1_kernel<<<blocks, 128, 0, stream>>>(x, W1, b1, h0);
    }
    // SpMM1: h1 += A @ h0
    {
        size_t threads = (size_t)N_EDGES * N_HID;
        int blocks = (int)((threads + 255) / 256);
        gcn_spmm64_kernel<<<blocks, 256, 0, stream>>>(adj_row, adj_col, adj_vals, h0, h1);
    }
    // GEMM2 (relu fused, WMMA f32)
    {
        int blocks = (N_TILES + 3) / 4;
        gcn_gemm2_kernel<<<blocks, 128, 0, stream>>>(h1, W2, b2, h2);
    }
    // SpMM2: s2 += A @ h2
    {
        size_t threads = (size_t)N_EDGES * N_CLS;
        int blocks = (int)((threads + 255) / 256);
        gcn_spmm16_kernel<<<blocks, 256, 0, stream>>>(adj_row, adj_col, adj_vals, h2, s2);
    }
    // log_softmax
    {
        int blocks = (N_NODES + 255) / 256;
        gcn_lsm_kernel<<<blocks, 256, 0, stream>>>(s2, out);
    }
}